// GNN_Model_72911364817693
// MI455X (gfx1250) — compile-verified
//
#include <hip/hip_runtime.h>
#include <hip/hip_bf16.h>
#include <math.h>

typedef __attribute__((ext_vector_type(2))) float v2f;
typedef __attribute__((ext_vector_type(8))) float v8f;

#define NND   128
#define E0E   16256      // 128*127 edges per graph
#define HID   32
#define MSTR  129        // padded adjacency stride (odd -> bank-conflict free)
#define KB    256        // K*BATCH graphs
#define BATCH 64
#define KS    4

__device__ __forceinline__ v8f wmma4(v2f a, v2f b, v8f c) {
  // D = A(16x4 f32) x B(4x16 f32) + C(16x16 f32)
  return __builtin_amdgcn_wmma_f32_16x16x4_f32(
      /*neg_a=*/false, a, /*neg_b=*/false, b,
      /*c_mod=*/(short)0, c, /*reuse_a=*/false, /*reuse_b=*/false);
}

// ---------------------------------------------------------------------------
// Kernel 1: one workgroup per graph. Full fused pipeline up to masked r_y.
// LDS layout (floats):
//   M   [128*129]   normalized adjacency, transposed: M[d][s]
//   Ha  [128*32]    feature buffer A (layer I/O)
//   Hb  [128*32]    feature buffer B (hop ping-pong, later yw)
//   W2l [6144]      W2 (192x32); reused as yT[32*128] at the end
//   Wl  [768]       W1 padded to 6 blocks of 4x32 (row 3 zero)
//   wwl [1024]      w_w (32x32)
//   b1s/b2s/bpl [32 each], dis/a0s/pth [128 each]
// ---------------------------------------------------------------------------
__global__ __launch_bounds__(256) void gnn_graph_kernel(
    const float* __restrict__ x,  const float* __restrict__ ew,
    const float* __restrict__ W1, const float* __restrict__ b1,
    const float* __restrict__ W2, const float* __restrict__ b2,
    const float* __restrict__ bpw, const float* __restrict__ ww,
    float* __restrict__ out_ry, float* __restrict__ out_aik)
{
  extern __shared__ float sm[];
  float* M   = sm;                 // 16512
  float* Ha  = M   + NND * MSTR;   // 4096
  float* Hb  = Ha  + NND * HID;    // 4096
  float* W2l = Hb  + NND * HID;    // 6144
  float* Wl  = W2l + 6144;         // 768
  float* wwl = Wl  + 768;          // 1024
  float* b1s = wwl + 1024;         // 32
  float* b2s = b1s + 32;           // 32
  float* bpl = b2s + 32;           // 32
  float* dis = bpl + 32;           // 128
  float* a0s = dis + 128;          // 128
  float* pth = a0s + 128;          // 128

  const int g    = blockIdx.x;
  const int tid  = threadIdx.x;
  const int wave = tid >> 5;       // 8 waves
  const int lane = tid & 31;
  const int l15  = lane & 15;
  const int lhi  = lane >> 4;      // 0|1 -> K-half / row-half select

  // ---- load raw adjacency (transposed) from flat edge list ----
  const float* ewg = ew + (size_t)g * E0E;
  for (int e = tid; e < E0E; e += 256) {
    int s = e / 127;
    int r = e - s * 127;
    int d = (r < s) ? r : r + 1;
    M[d * MSTR + s] = ewg[e];
  }
  if (tid < NND) M[tid * MSTR + tid] = 0.0f;

  // ---- features (zero-padded to 32 cols) + all weights into LDS ----
  for (int j = tid; j < NND * HID; j += 256) Ha[j] = 0.0f;
  if (tid < NND) {
    const float* xr = x + (size_t)(g * NND + tid) * 3;
    float a0 = xr[0], mu = xr[1], p = xr[2];
    a0s[tid] = a0; pth[tid] = p;
    Ha[tid * HID + 0] = a0;
    Ha[tid * HID + 1] = mu;
    Ha[tid * HID + 2] = p;
  }
  for (int j = tid; j < 768; j += 256) {                 // W1: pad 3->4 rows/block
    int k = j >> 7; int rr = (j >> 5) & 3; int c = j & 31;
    Wl[j] = (rr < 3) ? W1[(k * 3 + rr) * HID + c] : 0.0f;
  }
  for (int j = tid; j < 6144; j += 256) W2l[j] = W2[j];
  for (int j = tid; j < 1024; j += 256) wwl[j] = ww[j];
  if (tid < 32) { b1s[tid] = b1[tid]; b2s[tid] = b2[tid]; bpl[tid] = bpw[tid]; }
  __syncthreads();

  // ---- GCN norm: deg = row-sum of M; M[d][s] *= dis[s]*dis[d] ----
  if (tid < NND) {
    float s = 0.f;
    for (int j = 0; j < NND; ++j) s += M[tid * MSTR + j];
    dis[tid] = (s > 0.f) ? rsqrtf(s) : 0.f;
  }
  __syncthreads();
  for (int j = tid; j < NND * NND; j += 256) {
    int d = j >> 7, s = j & 127;
    M[d * MSTR + s] *= dis[s] * dis[d];
  }
  __syncthreads();

  // ---- hop: Hdst = M @ Hsrc (128x128 @ 128x32), wave -> row tile ----
  auto propagate = [&](const float* Hsrc, float* Hdst) {
    int arow = wave * 16 + l15;
    for (int nt = 0; nt < 2; ++nt) {
      v8f acc = {};
      int col = nt * 16 + l15;
      for (int k0 = 0; k0 < NND; k0 += 4) {
        int ka = k0 + (lhi << 1);
        v2f a; a.x = M[arow * MSTR + ka];     a.y = M[arow * MSTR + ka + 1];
        v2f b; b.x = Hsrc[ka * HID + col];    b.y = Hsrc[(ka + 1) * HID + col];
        acc = wmma4(a, b, acc);
      }
      int rbase = wave * 16 + (lhi << 3);
      #pragma unroll
      for (int r = 0; r < 8; ++r) Hdst[(rbase + r) * HID + col] = acc[r];
    }
  };

  // ---- Yacc += Hsrc(:,0:Kdim) @ Wblk(Kdim x 32) ----
  auto accW = [&](const float* Hsrc, const float* Wblk, int Kdim, v8f* acc) {
    int arow = wave * 16 + l15;
    for (int nt = 0; nt < 2; ++nt) {
      int col = nt * 16 + l15;
      for (int k0 = 0; k0 < Kdim; k0 += 4) {
        int ka = k0 + (lhi << 1);
        v2f a; a.x = Hsrc[arow * HID + ka];   a.y = Hsrc[arow * HID + ka + 1];
        v2f b; b.x = Wblk[ka * HID + col];    b.y = Wblk[(ka + 1) * HID + col];
        acc[nt] = wmma4(a, b, acc[nt]);
      }
    }
  };

  // ---- one TAGConv layer: y = lrelu(concat(h0..h5) @ W + b) -> Ha ----
  auto tag_layer = [&](const float* Wbase, int blkRows, int Kdim, const float* bias) {
    v8f acc[2];
    float bv0 = bias[l15], bv1 = bias[16 + l15];
    #pragma unroll
    for (int r = 0; r < 8; ++r) { acc[0][r] = bv0; acc[1][r] = bv1; }
    const float* Hcur = Ha; float* Hoth = Hb;
    for (int k = 0; k < 6; ++k) {
      if (k > 0) {
        propagate(Hcur, Hoth);
        __syncthreads();
        float* t = (float*)Hcur; Hcur = Hoth; Hoth = t;
      }
      accW(Hcur, Wbase + k * blkRows * HID, Kdim, acc);
    }
    __syncthreads();                       // all reads of H buffers done
    for (int nt = 0; nt < 2; ++nt) {       // leaky-relu, store y -> Ha
      int col = nt * 16 + l15;
      int rbase = wave * 16 + (lhi << 3);
      #pragma unroll
      for (int r = 0; r < 8; ++r) {
        float v = acc[nt][r];
        Ha[(rbase + r) * HID + col] = (v > 0.f) ? v : 0.01f * v;
      }
    }
    __syncthreads();
  };

  tag_layer(Wl,  4,  4,  b1s);   // layer 1: Fin=3 padded to K=4
  tag_layer(W2l, 32, 32, b2s);   // layer 2: Fin=32

  // ---- a_ik = (a0 + relu(y @ bp_w)) * (1 - path) ----
  if (tid < NND) {
    float s = 0.f;
    for (int f = 0; f < HID; ++f) s += Ha[tid * HID + f] * bpl[f];
    s = fmaxf(s, 0.f);
    out_aik[g * NND + tid] = (a0s[tid] + s) * (1.0f - pth[tid]);
  }

  // ---- yw = y @ w_w  -> Hb ----
  {
    int arow = wave * 16 + l15;
    for (int nt = 0; nt < 2; ++nt) {
      v8f acc = {};
      int col = nt * 16 + l15;
      for (int k0 = 0; k0 < HID; k0 += 4) {
        int ka = k0 + (lhi << 1);
        v2f a; a.x = Ha[arow * HID + ka];  a.y = Ha[arow * HID + ka + 1];
        v2f b; b.x = wwl[ka * HID + col];  b.y = wwl[(ka + 1) * HID + col];
        acc = wmma4(a, b, acc);
      }
      int rbase = wave * 16 + (lhi << 3);
      #pragma unroll
      for (int r = 0; r < 8; ++r) Hb[(rbase + r) * HID + col] = acc[r];
    }
  }
  // yT[f][m] = y[m][f]  (overlay on W2l, no longer needed)
  float* yT = W2l;
  for (int j = tid; j < HID * NND; j += 256) {
    int f = j >> 7, m = j & 127;
    yT[j] = Ha[m * HID + f];
  }
  __syncthreads();

  // ---- r_y = yw @ y^T (128x128), row-mask where path==1, -> global ----
  {
    int arow = wave * 16 + l15;
    float* outg = out_ry + (size_t)g * NND * NND;
    for (int nt = 0; nt < 8; ++nt) {
      v8f acc = {};
      int col = nt * 16 + l15;
      for (int k0 = 0; k0 < HID; k0 += 4) {
        int ka = k0 + (lhi << 1);
        v2f a; a.x = Hb[arow * HID + ka];   a.y = Hb[arow * HID + ka + 1];
        v2f b; b.x = yT[ka * NND + col];    b.y = yT[(ka + 1) * NND + col];
        acc = wmma4(a, b, acc);
      }
      int rbase = wave * 16 + (lhi << 3);
      #pragma unroll
      for (int r = 0; r < 8; ++r) {
        int row = rbase + r;
        float v = acc[r];
        if (pth[row] == 1.0f) v = -1e10f;
        outg[row * NND + col] = v;
      }
    }
  }
}

// ---------------------------------------------------------------------------
// Kernel 2: softmax over K=4 samples (in place on r_ij region) and * C,
// where C[kb][n][m] = eigen[kb] * ew(kb, edge n->m), diag 0.
// ---------------------------------------------------------------------------
__global__ __launch_bounds__(128) void softmax_mix_kernel(
    const float* __restrict__ ew, const float* __restrict__ eigen,
    float* __restrict__ rij)
{
  const int b = blockIdx.x;        // 0..63
  const int n = blockIdx.y;        // 0..127
  const int m = threadIdx.x;       // 0..127
  float v[KS];
  float mx = -INFINITY;
  #pragma unroll
  for (int k = 0; k < KS; ++k) {
    int kb = k * BATCH + b;
    v[k] = rij[(size_t)kb * (NND * NND) + n * NND + m];
    mx = fmaxf(mx, v[k]);
  }
  float s = 0.f;
  #pragma unroll
  for (int k = 0; k < KS; ++k) { v[k] = __expf(v[k] - mx); s += v[k]; }
  float inv = 1.0f / s;
  #pragma unroll
  for (int k = 0; k < KS; ++k) {
    int kb = k * BATCH + b;
    float c = 0.f;
    if (m != n) {
      int eidx = n * 127 + ((m < n) ? m : m - 1);
      c = ew[(size_t)kb * E0E + eidx] * eigen[kb];
    }
    rij[(size_t)kb * (NND * NND) + n * NND + m] = v[k] * inv * c;
  }
}

// ---------------------------------------------------------------------------
// Kernel 3: per-graph 128x128 transpose (coalesced both ways via padded LDS)
// ---------------------------------------------------------------------------
__global__ __launch_bounds__(256) void transpose_kernel(
    const float* __restrict__ rij, float* __restrict__ rji)
{
  extern __shared__ float t[];     // 128*129
  const int g = blockIdx.x;
  const float* src = rij + (size_t)g * (NND * NND);
  float* dst = rji + (size_t)g * (NND * NND);
  for (int j = threadIdx.x; j < NND * NND; j += 256) {
    int n = j >> 7, m = j & 127;
    t[n * MSTR + m] = src[j];
  }
  __syncthreads();
  for (int j = threadIdx.x; j < NND * NND; j += 256) {
    int n = j >> 7, m = j & 127;
    dst[j] = t[m * MSTR + n];      // odd stride -> no bank conflict
  }
}

extern "C" void kernel_launch(void* const* d_in, const int* in_sizes, int n_in,
                              void* d_out, int out_size, void* d_ws, size_t ws_size,
                              hipStream_t stream) {
  (void)in_sizes; (void)n_in; (void)out_size; (void)d_ws; (void)ws_size;
  const float* x     = (const float*)d_in[0];
  // d_in[1] = edge_index: structure is known analytically (complete digraph)
  const float* ew    = (const float*)d_in[2];
  const float* eigen = (const float*)d_in[3];
  const float* W1    = (const float*)d_in[4];
  const float* b1    = (const float*)d_in[5];
  const float* W2    = (const float*)d_in[6];
  const float* b2    = (const float*)d_in[7];
  const float* bpw   = (const float*)d_in[8];
  const float* ww    = (const float*)d_in[9];

  float* out = (float*)d_out;
  float* rij = out;                                   // KB*N*N
  float* rji = out + (size_t)KB * NND * NND;          // KB*N*N
  float* aik = out + (size_t)2 * KB * NND * NND;      // KB*N

  const size_t lds1 = (size_t)(NND * MSTR + 2 * NND * HID + 6144 + 768 + 1024
                               + 3 * 32 + 3 * 128) * sizeof(float);   // ~129.4 KB
  gnn_graph_kernel<<<KB, 256, lds1, stream>>>(x, ew, W1, b1, W2, b2, bpw, ww,
                                              rij, aik);

  dim3 g2(BATCH, NND);
  softmax_mix_kernel<<<g2, NND, 0, stream>>>(ew, eigen, rij);

  transpose_kernel<<<KB, 256, (size_t)NND * MSTR * sizeof(float), stream>>>(rij, rji);
}